// LeakyAverageCuda_34385508172337
// MI455X (gfx1250) — compile-verified
//
#include <hip/hip_runtime.h>
#include <stdint.h>

#ifndef __has_builtin
#define __has_builtin(x) 0
#endif

#if __has_builtin(__builtin_amdgcn_global_load_async_to_lds_b128)
#define HAVE_ASYNC_LDS 1
#else
#define HAVE_ASYNC_LDS 0
#endif

// Problem dims (fixed by the reference)
#define NB 16
#define NT 1024
#define NC 2048
#define NH 16
#define HDIM (NC / NH)   // 128 channels per head

// Pipeline configuration
#define DTILE 32                          // timesteps staged per tile
#define NTILES (NT / DTILE)               // 32
#define NBUF 3                            // triple buffer
#define WAVES_PER_BLOCK 2
#define BLOCK_SIZE (WAVES_PER_BLOCK * 32) // 64 threads
#define ASYNC_PER_TILE ((DTILE / 4) * 2)  // 16 b128 asyncs per tile (k + v)

#define LOG2E_F 1.4426950408889634f

// The async-to-LDS builtin takes typed int4 pointers:
//   arg0: v4i addrspace(1)* (global source)   -- printed as "int __vector(4) __device__ *"
//   arg1: v4i addrspace(3)* (LDS destination)
typedef int v4i __attribute__((ext_vector_type(4)));
typedef __attribute__((address_space(1))) v4i GV4;
typedef __attribute__((address_space(3))) v4i LV4;

__device__ __forceinline__ GV4* as_global(const void* p) {
  // generic -> global: same bits; int round-trip avoids addrspace-cast rules
  return (GV4*)(unsigned long long)p;
}
__device__ __forceinline__ LV4* as_lds(const void* p) {
  // generic shared pointer: low 32 bits are the LDS byte offset
  return (LV4*)(unsigned)(unsigned long long)p;
}

__device__ __forceinline__ float fast_exp2(float x) {
#if __has_builtin(__builtin_amdgcn_exp2f)
  return __builtin_amdgcn_exp2f(x);   // v_exp_f32 (2^x)
#else
  return exp2f(x);
#endif
}
__device__ __forceinline__ float fast_rcp(float x) {
#if __has_builtin(__builtin_amdgcn_rcpf)
  return __builtin_amdgcn_rcpf(x);    // v_rcp_f32
#else
  return 1.0f / x;
#endif
}

template <int N>
__device__ __forceinline__ void wait_asynccnt_le() {
#if __has_builtin(__builtin_amdgcn_s_wait_asynccnt)
  __builtin_amdgcn_s_wait_asynccnt(N);
#else
  asm volatile("s_wait_asynccnt %0" ::"i"(N) : "memory");
#endif
  asm volatile("" ::: "memory");  // compiler fence: no LDS reads hoisted above
}

__global__ __launch_bounds__(BLOCK_SIZE)
void wkv_scan_kernel(const float* __restrict__ v_x,     // v  = x     [B,T,C]
                     const float* __restrict__ k_gate,  // k  = gate  [B,T,C]
                     const float* __restrict__ beta_w,  // [H]
                     const float* __restrict__ beta_u,  // [H]
                     float* __restrict__ y)             // [B,T,C]
{
  // LDS staging: per wave, NBUF buffers of DTILE x 32 floats for k and v.
  __shared__ float sK[WAVES_PER_BLOCK][NBUF][DTILE][32];
  __shared__ float sV[WAVES_PER_BLOCK][NBUF][DTILE][32];

  const int lane = threadIdx.x & 31;
  const int wv   = threadIdx.x >> 5;
  const int g    = blockIdx.x * BLOCK_SIZE + threadIdx.x;  // 0 .. B*C-1
  const int b    = g / NC;
  const int c    = g % NC;
  const int h    = c / HDIM;                 // wave-uniform (32 chans < HDIM=128)

  // Per-channel params, pre-scaled to base-2 exponent space.
  const float w2 = -fabsf(beta_w[h]) * 10.0f * LOG2E_F;  // negative decay
  const float u2 =  fabsf(beta_u[h]) * 10.0f * LOG2E_F;  // bonus

  const int    cwave  = c & ~31;                          // wave's base channel
  const size_t baseBC = (size_t)b * NT * NC;

  // Async lane mapping: one b128 per lane = 4 channels of one timestep.
  // Lanes 0-7 -> t-subrow 0, lanes 8-15 -> subrow 1, etc.
  const int tsub = lane >> 3;        // 0..3
  const int csub = (lane & 7) * 4;   // 0,4,...,28

  float p = 0.0f, q = 0.0f, o = -1.0e38f;
  const float* kcol = k_gate + baseBC + c;   // per-step scalar fallback column
  const float* vcol = v_x    + baseBC + c;
  float*       ycol = y      + baseBC + c;

#if HAVE_ASYNC_LDS
  const float* kp0 = k_gate + baseBC + cwave + csub;   // per-lane async source base
  const float* vp0 = v_x    + baseBC + cwave + csub;

  auto issue_tile = [&](int tile, int buf) {
    const int t0 = tile * DTILE;
#pragma unroll
    for (int a = 0; a < DTILE / 4; ++a) {
      const int trow = a * 4 + tsub;                 // 0..DTILE-1
      const size_t goff = (size_t)(t0 + trow) * NC;  // floats
      __builtin_amdgcn_global_load_async_to_lds_b128(
          as_global(kp0 + goff), as_lds(&sK[wv][buf][trow][csub]), 0, 0);
      __builtin_amdgcn_global_load_async_to_lds_b128(
          as_global(vp0 + goff), as_lds(&sV[wv][buf][trow][csub]), 0, 0);
    }
  };

  // Prologue: fill the pipeline with 3 tiles (48 asyncs, ASYNCcnt max 63).
  issue_tile(0, 0);
  issue_tile(1, 1);
  issue_tile(2, 2);

  for (int tile = 0; tile < NTILES; ++tile) {
    const int buf = tile % NBUF;

    // In-order completion: wait until only the not-yet-needed tiles remain.
    if (tile + 2 < NTILES)       wait_asynccnt_le<2 * ASYNC_PER_TILE>();
    else if (tile + 1 < NTILES)  wait_asynccnt_le<ASYNC_PER_TILE>();
    else                         wait_asynccnt_le<0>();

    const int t0 = tile * DTILE;
#pragma unroll 8
    for (int tt = 0; tt < DTILE; ++tt) {
      const float kc = sK[wv][buf][tt][lane] * LOG2E_F;  // base-2 exponent space
      const float vv = sV[wv][buf][tt][lane];

      // Output at t (includes u-bonus on current token)
      const float uk  = u2 + kc;
      const float no  = fmaxf(o, uk);
      const float A   = fast_exp2(o - no);
      const float Bc  = fast_exp2(uk - no);
      const float num = A * p + Bc * vv;
      const float den = A * q + Bc;
      const float yv  = num * fast_rcp(den);
      __builtin_nontemporal_store(yv, ycol + (size_t)(t0 + tt) * NC);

      // State update (decay, no bonus)
      const float ow  = o + w2;
      const float no2 = fmaxf(ow, kc);
      const float A2  = fast_exp2(ow - no2);
      const float B2  = fast_exp2(kc - no2);
      p = A2 * p + B2 * vv;
      q = A2 * q + B2;
      o = no2;
    }

    asm volatile("" ::: "memory");  // reads of `buf` done before overwrite
    if (tile + NBUF < NTILES) issue_tile(tile + NBUF, buf);
  }
#else
  // Fallback: direct streaming loop with software prefetch.
  for (int t = 0; t < NT; ++t) {
    if (t + 8 < NT) {
      __builtin_prefetch(kcol + (size_t)(t + 8) * NC, 0, 0);
      __builtin_prefetch(vcol + (size_t)(t + 8) * NC, 0, 0);
    }
    const float kc = __builtin_nontemporal_load(kcol + (size_t)t * NC) * LOG2E_F;
    const float vv = __builtin_nontemporal_load(vcol + (size_t)t * NC);

    const float uk  = u2 + kc;
    const float no  = fmaxf(o, uk);
    const float A   = fast_exp2(o - no);
    const float Bc  = fast_exp2(uk - no);
    const float yv  = (A * p + Bc * vv) * fast_rcp(A * q + Bc);
    __builtin_nontemporal_store(yv, ycol + (size_t)t * NC);

    const float ow  = o + w2;
    const float no2 = fmaxf(ow, kc);
    const float A2  = fast_exp2(ow - no2);
    const float B2  = fast_exp2(kc - no2);
    p = A2 * p + B2 * vv;
    q = A2 * q + B2;
    o = no2;
  }
#endif
}

extern "C" void kernel_launch(void* const* d_in, const int* in_sizes, int n_in,
                              void* d_out, int out_size, void* d_ws, size_t ws_size,
                              hipStream_t stream) {
  (void)in_sizes; (void)n_in; (void)out_size; (void)d_ws; (void)ws_size;
  const float* x      = (const float*)d_in[0];  // v
  const float* gate   = (const float*)d_in[1];  // k
  const float* beta_w = (const float*)d_in[2];  // [H]
  const float* beta_u = (const float*)d_in[3];  // [H]
  float* yout = (float*)d_out;

  const int total_threads = NB * NC;            // one thread per (b, c) scan
  dim3 grid(total_threads / BLOCK_SIZE);        // 512 blocks
  dim3 block(BLOCK_SIZE);                       // 64 threads = 2 waves
  wkv_scan_kernel<<<grid, block, 0, stream>>>(x, gate, beta_w, beta_u, yout);
}